// GCN_classifier_1202590843008
// MI455X (gfx1250) — compile-verified
//
#include <hip/hip_runtime.h>
#include <cstddef>

// CDNA5 wave32 WMMA fragments
typedef float v2f __attribute__((ext_vector_type(2)));
typedef float v8f __attribute__((ext_vector_type(8)));

#define HIDC 128
#define NCLS 16

// ---------------- degree / normalization ----------------
__global__ void k_init_deg(float* __restrict__ deg, int n) {
  int i = blockIdx.x * blockDim.x + threadIdx.x;
  if (i < n) deg[i] = 1.0f;  // self-loop contributes 1
}

__global__ void k_deg_accum(const int* __restrict__ dst, float* __restrict__ deg, int e) {
  int i = blockIdx.x * blockDim.x + threadIdx.x;
  if (i < e) atomicAdd(&deg[dst[i]], 1.0f);
}

__global__ void k_rsqrt_inplace(float* __restrict__ deg, int n) {
  int i = blockIdx.x * blockDim.x + threadIdx.x;
  if (i < n) {
    float d = deg[i];
    deg[i] = (d > 0.0f) ? rsqrtf(d) : 0.0f;
  }
}

// ---------------- zero fill ----------------
__global__ void k_zero(float* __restrict__ p, size_t n) {
  size_t i = (size_t)blockIdx.x * blockDim.x + threadIdx.x;
  if (i < n) p[i] = 0.0f;
}

// ---------------- fp32 WMMA GEMM: C[N x 128] = A[N x 128] @ W[128 x 128] ----------------
// Block = 256 threads = 8 wave32 waves. Block owns 16 rows; wave w owns cols [16w, 16w+16).
// V_WMMA_F32_16X16X4_F32: A 16x4 (v2f/lane), B 4x16 (v2f/lane), C/D 16x16 f32 (v8f/lane).
// A layout: lanes 0-15 hold M=lane, {K=k0,k0+1}; lanes 16-31 hold M=lane-16, {K=k0+2,k0+3}.
// B layout (mirror): lanes 0-15 hold N=lane rows {k0,k0+1}; lanes 16-31 rows {k0+2,k0+3}.
// C/D layout: VGPR j = row (j + 8*half), col = lane&15.
__global__ __launch_bounds__(256) void k_gemm_wmma_f32(
    const float* __restrict__ A, const float* __restrict__ W,
    float* __restrict__ C, int nrows) {
  const int wave = threadIdx.x >> 5;     // 0..7 -> column tile
  const int lane = threadIdx.x & 31;
  const int half = lane >> 4;            // 0 or 1
  const int m    = lane & 15;
  const int r0   = blockIdx.x * 16;
  const int n0   = wave * 16;

  int arow = r0 + m;
  if (arow >= nrows) arow = nrows - 1;   // clamp (never taken when N%16==0)
  const float* __restrict__ ap = A + (size_t)arow * HIDC;

  v8f acc = {};
  #pragma unroll 8
  for (int k0 = 0; k0 < HIDC; k0 += 4) {
    const int k = k0 + (half << 1);
    v2f a; a.x = ap[k];
           a.y = ap[k + 1];
    v2f b; b.x = W[(size_t)k * HIDC + n0 + m];
           b.y = W[(size_t)(k + 1) * HIDC + n0 + m];
    acc = __builtin_amdgcn_wmma_f32_16x16x4_f32(
        /*neg_a=*/false, a, /*neg_b=*/false, b,
        /*c_mod=*/(short)0, acc, /*reuse_a=*/false, /*reuse_b=*/false);
  }

  const int rbase = r0 + half * 8;
  float* __restrict__ cp = C + (size_t)rbase * HIDC + n0 + m;
  if (r0 + 16 <= nrows) {
    // uniform fast path: whole 16-row strip in range -> straight-line stores
    #pragma unroll
    for (int j = 0; j < 8; ++j) cp[(size_t)j * HIDC] = acc[j];
  } else {
    #pragma unroll
    for (int j = 0; j < 8; ++j) {
      if (rbase + j < nrows) cp[(size_t)j * HIDC] = acc[j];
    }
  }
}

// ---------------- edge gather/scale/scatter-add ----------------
// One wave per edge: 32 lanes x float4 = 128 channels. agg[dst] += h[src] * dinv[s]*dinv[d].
__global__ __launch_bounds__(256) void k_scatter(
    const float* __restrict__ h, const int* __restrict__ src,
    const int* __restrict__ dst, const float* __restrict__ dinv,
    float* __restrict__ agg, int e) {
  size_t tid = (size_t)blockIdx.x * blockDim.x + threadIdx.x;
  int edge = (int)(tid >> 5);
  int lane = (int)(tid & 31);
  if (edge >= e) return;
  int s = src[edge];
  int d = dst[edge];
  float w = dinv[s] * dinv[d];
  const float4* __restrict__ hv = (const float4*)(h + (size_t)s * HIDC);
  float4 v = hv[lane];
  float* __restrict__ o = agg + (size_t)d * HIDC + lane * 4;
  atomicAdd(o + 0, v.x * w);
  atomicAdd(o + 1, v.y * w);
  atomicAdd(o + 2, v.z * w);
  atomicAdd(o + 3, v.w * w);
}

// ---------------- self-loop + bias + ReLU (fused epilogue) ----------------
__global__ void k_selfloop_bias_relu(
    float* __restrict__ agg, const float* __restrict__ h,
    const float* __restrict__ dinv, const float* __restrict__ bias, int n) {
  size_t tid = (size_t)blockIdx.x * blockDim.x + threadIdx.x;
  if (tid >= (size_t)n * HIDC) return;
  int node = (int)(tid >> 7);
  int c    = (int)(tid & (HIDC - 1));
  float di = dinv[node];
  float v = agg[tid] + h[tid] * di * di + bias[c];
  agg[tid] = fmaxf(v, 0.0f);
}

// ---------------- classifier head: logits = h @ Wl + bl, softmax over 16 ----------------
// 16 threads per node; class-c logit per thread; shuffle reductions within the 16-group.
__global__ void k_logits_softmax(
    const float* __restrict__ h, const float* __restrict__ Wl,
    const float* __restrict__ bl, float* __restrict__ out, int n) {
  size_t tid = (size_t)blockIdx.x * blockDim.x + threadIdx.x;
  int node = (int)(tid >> 4);
  int c    = (int)(tid & 15);
  if (node >= n) return;
  const float* __restrict__ hr = h + (size_t)node * HIDC;
  float acc = bl[c];
  #pragma unroll 8
  for (int k = 0; k < HIDC; ++k)
    acc = fmaf(hr[k], Wl[k * NCLS + c], acc);
  float mx = acc;
  for (int off = 8; off > 0; off >>= 1)
    mx = fmaxf(mx, __shfl_xor(mx, off, 16));
  float ex = __expf(acc - mx);
  float sum = ex;
  for (int off = 8; off > 0; off >>= 1)
    sum += __shfl_xor(sum, off, 16);
  out[(size_t)node * NCLS + c] = ex / sum;
}

extern "C" void kernel_launch(void* const* d_in, const int* in_sizes, int n_in,
                              void* d_out, int out_size, void* d_ws, size_t ws_size,
                              hipStream_t stream) {
  (void)n_in; (void)out_size; (void)ws_size;

  const float* x  = (const float*)d_in[0];
  const int*   ei = (const int*)d_in[1];   // [2, E]: row0 = src, row1 = dst
  const float* W1 = (const float*)d_in[2];
  const float* b1 = (const float*)d_in[3];
  const float* W2 = (const float*)d_in[4];
  const float* b2 = (const float*)d_in[5];
  const float* Wl = (const float*)d_in[6];
  const float* bl = (const float*)d_in[7];
  float* out = (float*)d_out;

  const int N = in_sizes[0] / HIDC;
  const int E = in_sizes[1] / 2;
  const int* src = ei;
  const int* dst = ei + E;

  // workspace: dinv[N] | h[N*128] | agg[N*128]  (~103 MB, L2-resident working set)
  float* dinv = (float*)d_ws;
  float* hbuf = dinv + N;
  float* abuf = hbuf + (size_t)N * HIDC;

  const int T = 256;
  const size_t nh = (size_t)N * HIDC;
  const int gN    = (N + T - 1) / T;
  const int gE    = (E + T - 1) / T;
  const int gNH   = (int)((nh + T - 1) / T);
  const int gEdge = (int)(((size_t)E * 32 + T - 1) / T);
  const int gGemm = (N + 15) / 16;
  const int gSoft = (int)(((size_t)N * NCLS + T - 1) / T);

  // symmetric normalization: deg (incl. self-loop) -> dinv = rsqrt(deg)
  k_init_deg<<<gN, T, 0, stream>>>(dinv, N);
  k_deg_accum<<<gE, T, 0, stream>>>(dst, dinv, E);
  k_rsqrt_inplace<<<gN, T, 0, stream>>>(dinv, N);

  // layer 1: h = x @ W1 ; agg = scatter(norm * h[src]) + selfloop ; relu(+b1)
  k_gemm_wmma_f32<<<gGemm, T, 0, stream>>>(x, W1, hbuf, N);
  k_zero<<<gNH, T, 0, stream>>>(abuf, nh);
  k_scatter<<<gEdge, T, 0, stream>>>(hbuf, src, dst, dinv, abuf, E);
  k_selfloop_bias_relu<<<gNH, T, 0, stream>>>(abuf, hbuf, dinv, b1, N);

  // layer 2
  k_gemm_wmma_f32<<<gGemm, T, 0, stream>>>(abuf, W2, hbuf, N);
  k_zero<<<gNH, T, 0, stream>>>(abuf, nh);
  k_scatter<<<gEdge, T, 0, stream>>>(hbuf, src, dst, dinv, abuf, E);
  k_selfloop_bias_relu<<<gNH, T, 0, stream>>>(abuf, hbuf, dinv, b2, N);

  // classifier head + softmax
  k_logits_softmax<<<gSoft, T, 0, stream>>>(abuf, Wl, bl, out, N);
}